// PercolationC_31885837205968
// MI455X (gfx1250) — compile-verified
//
#include <hip/hip_runtime.h>
#include <stdint.h>

// ---------------------------------------------------------------------------
// Connected-component counting on 2048 independent 64x64 binary images,
// then integer-mean over groups of 64 -> 32 floats.
//
// CDNA5 specifics used:
//  * TDM tensor_load_to_lds (D# built per CDNA5 ISA)
//    to DMA each 16 KiB f32 tile into LDS; completion via s_wait_tensorcnt.
//  * In-LDS min-label propagation with pointer jumping (ds_load/ds_store),
//    LDS atomics (ds_add) for root counting; wave32 / 8-wave workgroups.
//
// Toolchain note: this compiler exposes the 6-arg form of
// __builtin_amdgcn_tensor_load_to_lds:
//   (uint32x4 g0, int32x8 g1, int32x4 g2, int32x4 g3, int32x8 extra, i32 cpol)
// ---------------------------------------------------------------------------

#define BIGL 0x40000000u

typedef unsigned int u32;
typedef __attribute__((ext_vector_type(4))) u32 v4u;
typedef __attribute__((ext_vector_type(8))) int v8i;
typedef __attribute__((ext_vector_type(4))) int v4i;

static __device__ __forceinline__ u32 umin32(u32 a, u32 b) { return a < b ? a : b; }

// Build a Tensor DMA Descriptor for a 64x64 f32 tile (tensor == tile here)
// and issue TENSOR_LOAD_TO_LDS.  Layout per ISA §8.3/§8.4:
//   group0: [1:0]=count=1, [63:32]=lds_addr, [120:64]=global_addr, [127:126]=type=2
//   group1: [17:16]=data_size(2 -> 4B), [79:48]=tensor_dim0=64, [111:80]=tensor_dim1=64,
//           [127:112]=tile_dim0=64, [143:128]=tile_dim1=64, [207:160]=dim0_stride=64,
//           [255:208]=dim1_stride=4096 (unused for 2D tile)
//   group2/3 and extra group: zero (dims 2..4 unused)
static __device__ __forceinline__ void tdm_load_tile_f32_64x64(const void* gptr,
                                                               u32 lds_byte_off) {
  unsigned long long ga = (unsigned long long)gptr;
  v4u g0;
  g0.x = 1u;                                              // count=1 (valid, user mode)
  g0.y = lds_byte_off;                                    // lds_addr (bytes)
  g0.z = (u32)(ga & 0xFFFFFFFFull);                       // global_addr[31:0]
  g0.w = (u32)((ga >> 32) & 0x01FFFFFFull) | (2u << 30);  // global_addr[56:32] | type=2
  v8i g1;
  g1[0] = (int)(2u << 16);      // workgroup_mask=0, data_size=2 (4 bytes)
  g1[1] = (int)(64u << 16);     // tensor_dim0[15:0] = 64  (bits 63:48)
  g1[2] = (int)(64u << 16);     // tensor_dim0 hi=0, tensor_dim1[15:0]=64 (bits 95:80)
  g1[3] = (int)(64u << 16);     // tensor_dim1 hi=0, tile_dim0=64 (bits 127:112)
  g1[4] = 64;                   // tile_dim1=64 (bits 143:128), tile_dim2=0
  g1[5] = 64;                   // tensor_dim0_stride[31:0] = 64 elements
  g1[6] = (int)(4096u << 16);   // stride0 hi=0, tensor_dim1_stride[15:0]=4096
  g1[7] = 0;                    // tensor_dim1_stride hi
  v4i g2 = {0, 0, 0, 0};
  v4i g3 = {0, 0, 0, 0};
  v8i gx = {0, 0, 0, 0, 0, 0, 0, 0};
  __builtin_amdgcn_tensor_load_to_lds(g0, g1, g2, g3, gx, 0);
}

// One block per 64x64 image.  256 threads = 8 wave32.
__global__ void __launch_bounds__(256) cc_count_kernel(const float* __restrict__ x,
                                                       int* __restrict__ counts) {
  __shared__ u32 buf[4096];   // TDM target; reused in-place as the label array
  __shared__ int changed;
  __shared__ int cnt;
  const int tid = threadIdx.x;

  // --- Stage the tile into LDS via the Tensor Data Mover (one wave issues) ---
  if (tid == 0) {
    u32 lds_off = (u32)(uintptr_t)(void*)&buf[0];  // flat LDS addr low 32b = LDS offset
    tdm_load_tile_f32_64x64(x + (size_t)blockIdx.x * 4096u, lds_off);
    __builtin_amdgcn_s_wait_tensorcnt(0);
  }
  __syncthreads();

  // --- In-place convert f32 pixel -> initial label (own index, or BIG for bg) ---
  for (int i = tid; i < 4096; i += 256) {
    float v = __uint_as_float(buf[i]);
    buf[i] = (v > 0.5f) ? (u32)i : BIGL;
  }

  // --- Min-label propagation with double pointer jumping until fixed point ---
  for (;;) {
    __syncthreads();               // separates previous flag read from reset
    if (tid == 0) changed = 0;
    __syncthreads();
    int any = 0;
    for (int i = tid; i < 4096; i += 256) {
      u32 l = buf[i];
      if (l >= BIGL) continue;     // background
      u32 m = l;
      const int y = i >> 6;
      const int c = i & 63;
      if (y > 0)  m = umin32(m, buf[i - 64]);
      if (y < 63) m = umin32(m, buf[i + 64]);
      if (c > 0)  m = umin32(m, buf[i - 1]);
      if (c < 63) m = umin32(m, buf[i + 1]);
      // m is always a foreground pixel index (< 4096): safe to chase.
      m = umin32(m, buf[m]);
      m = umin32(m, buf[m]);
      if (m < l) { buf[i] = m; any = 1; }
    }
    if (any) changed = 1;          // benign racy same-value store
    __syncthreads();
    if (changed == 0) break;       // a full pass with zero updates == fixed point
  }

  // --- Count roots: exactly one pixel per component keeps lbl[i]==i ---
  if (tid == 0) cnt = 0;
  __syncthreads();
  int c = 0;
  for (int i = tid; i < 4096; i += 256) c += (buf[i] == (u32)i) ? 1 : 0;
  atomicAdd(&cnt, c);              // LDS ds_add
  __syncthreads();
  if (tid == 0) counts[blockIdx.x] = cnt;
}

// Integer-truncating mean over each group of 64 patch counts.
__global__ void group_reduce_kernel(const int* __restrict__ counts,
                                    float* __restrict__ out, int ngroups) {
  int g = blockIdx.x * blockDim.x + threadIdx.x;
  if (g >= ngroups) return;
  const int* p = counts + g * 64;
  int s = 0;
#pragma unroll
  for (int i = 0; i < 64; ++i) s += p[i];
  out[g] = (float)(s / 64);        // matches jnp integer // 64 (nonnegative)
}

extern "C" void kernel_launch(void* const* d_in, const int* in_sizes, int n_in,
                              void* d_out, int out_size, void* d_ws, size_t ws_size,
                              hipStream_t stream) {
  (void)n_in; (void)ws_size;
  const float* x = (const float*)d_in[0];
  float* out = (float*)d_out;
  int* counts = (int*)d_ws;                 // nimg ints of scratch (8 KiB)

  const int nimg = in_sizes[0] / 4096;      // 2048 images of 64x64
  const int ngroups = out_size;             // 32 groups of 64 patches

  cc_count_kernel<<<nimg, 256, 0, stream>>>(x, counts);
  group_reduce_kernel<<<(ngroups + 63) / 64, 64, 0, stream>>>(counts, out, ngroups);
}